// SkipGramModel_76536317215155
// MI455X (gfx1250) — compile-verified
//
#include <hip/hip_runtime.h>

// ---- problem constants (match reference) ----
#define EMB   500000
#define DIM   128
#define TOK   10240
#define NPOS  102400
#define NNEG  51200
#define LUT_N 1202
#define LR_C  0.025f

typedef __attribute__((ext_vector_type(2))) float v2f;
typedef __attribute__((ext_vector_type(8))) float v8f;

__device__ __forceinline__ float rdlane_f(float v, int l) {
  return __builtin_bit_cast(float, __builtin_amdgcn_readlane(__builtin_bit_cast(int, v), l));
}

// ---------------------------------------------------------------------------
// Phase 1: stream-copy both tables into d_out (new_u | new_v). 1 GB of HBM
// traffic -> this is the roofline cost (~45us @ 23.3 TB/s). float4 = b128 ops.
// ---------------------------------------------------------------------------
__global__ void copy_tables(const float4* __restrict__ su,
                            const float4* __restrict__ sv,
                            float4* __restrict__ dst) {
  const long N4 = (long)EMB * DIM / 4;                 // 16,000,000 per table
  const long stride = (long)gridDim.x * blockDim.x;
  for (long i = blockIdx.x * (long)blockDim.x + threadIdx.x; i < N4; i += stride) {
    dst[i]      = su[i];
    dst[N4 + i] = sv[i];
  }
}

// Phase 2: zero the per-token gradient buffers in workspace (deterministic:
// re-zeroed every call, so graph replays are self-consistent).
__global__ void zero_grads(float4* __restrict__ g) {
  const long N4 = (long)TOK * DIM * 2 / 4;             // 655,360
  long i = blockIdx.x * (long)blockDim.x + threadIdx.x;
  if (i < N4) g[i] = make_float4(0.f, 0.f, 0.f, 0.f);
}

// ---------------------------------------------------------------------------
// Phase 3: pair scores via V_WMMA_F32_16X16X4_F32 (exact f32), then LUT
// sigmoid and atomic rank-1 gradient scatter. One wave32 handles 16 pairs.
//
// A-matrix 16x4 f32 layout (ISA 7.12.2): lane L<16 holds row M=L, K={0,1};
// lane L+16 holds row M=L, K={2,3}.  B 4x16 is symmetric with N=lane%16.
// With A(i,k)=eu_i[k], B(k,j)=ev_j[k], accumulating 32 K-chunks gives
// C(i,j)=dot(eu_i, ev_j); we need only the diagonal.
// ---------------------------------------------------------------------------
__global__ void __launch_bounds__(256) pair_grads(
    const float* __restrict__ U, const float* __restrict__ V,
    const float* __restrict__ lut, const int* __restrict__ flat,
    const int* __restrict__ ipu, const int* __restrict__ ipv,
    const int* __restrict__ inu, const int* __restrict__ inv,
    float* __restrict__ gu, float* __restrict__ gv)
{
  const int NPG = NPOS / 16;                    // 6400 positive groups
  const int NNG = NNEG / 16;                    // 3200 negative groups
  const int wave = blockIdx.x * (blockDim.x >> 5) + (threadIdx.x >> 5);
  const int lane = threadIdx.x & 31;
  if (wave >= NPG + NNG) return;                // grid is exact; whole waves only

  const bool isneg = (wave >= NPG);
  const int grp = isneg ? (wave - NPG) : wave;
  const int* __restrict__ iu = isneg ? inu : ipu;
  const int* __restrict__ iv = isneg ? inv : ipv;

  // per-lane WMMA operand addressing
  const int  pr   = grp * 16 + (lane & 15);     // pair handled by this lane
  const int  half = lane >> 4;                  // K sub-block {0,1} vs {2,3}
  const float2* __restrict__ arow = (const float2*)(U + (long)flat[iu[pr]] * DIM);
  const float2* __restrict__ brow = (const float2*)(V + (long)flat[iv[pr]] * DIM);

  v8f c = {0.f, 0.f, 0.f, 0.f, 0.f, 0.f, 0.f, 0.f};
#pragma unroll
  for (int kk = 0; kk < 32; ++kk) {             // K = kk*4 + half*2 + {0,1}
    float2 af = arow[kk * 2 + half];
    float2 bf = brow[kk * 2 + half];
    v2f a; a.x = af.x; a.y = af.y;
    v2f b; b.x = bf.x; b.y = bf.y;
    c = __builtin_amdgcn_wmma_f32_16x16x4_f32(false, a, false, b, (short)0, c,
                                              false, false);
  }

  // Diagonal extraction (C layout: VGPR r = rows {r, r+8}; lanes 0-15 -> N,
  // lanes 16-31 -> N-16). diag(i): i<8 -> c[i]@lane i ; i>=8 -> c[i-8]@lane 24+(i-8).
  float sc[16];
#pragma unroll
  for (int i = 0; i < 8; ++i) {
    sc[i]     = rdlane_f(c[i], i);
    sc[i + 8] = rdlane_f(c[i], 24 + i);
  }

#pragma unroll
  for (int i = 0; i < 16; ++i) {
    // table sigmoid, same arithmetic as the reference
    float s  = fminf(6.0f, fmaxf(-6.0f, sc[i]));
    int   li = (int)floorf((s + 6.01f) / 0.01f);
    li = min(max(li, 0), LUT_N - 1);
    float sg   = lut[li];
    float coef = isneg ? (-sg) : (1.0f - sg);   // NEG_WEIGHT == 1.0

    const int ti_u = iu[grp * 16 + i];
    const int ti_v = iv[grp * 16 + i];
    const float* __restrict__ ur = U + (long)flat[ti_u] * DIM;
    const float* __restrict__ vr = V + (long)flat[ti_v] * DIM;
    float* __restrict__ gur = gu + (long)ti_u * DIM;
    float* __restrict__ gvr = gv + (long)ti_v * DIM;
#pragma unroll
    for (int d = 0; d < DIM; d += 32) {
      const int dd = d + lane;
      atomicAdd(&gur[dd], coef * vr[dd]);       // grad_u[tu] += g * ev
      atomicAdd(&gvr[dd], coef * ur[dd]);       // grad_v[tv] += g * eu
    }
  }
}

// ---------------------------------------------------------------------------
// Phase 4: out[flat[t]] += lr*0.025*grad[t]; atomics handle duplicate tokens
// (jax .at[].add semantics).
// ---------------------------------------------------------------------------
__global__ void apply_sgd(const int* __restrict__ flat,
                          const float* __restrict__ gu,
                          const float* __restrict__ gv,
                          const float* __restrict__ lr,
                          float* __restrict__ out_u, float* __restrict__ out_v)
{
  const long N = (long)TOK * DIM;
  long i = blockIdx.x * (long)blockDim.x + threadIdx.x;
  if (i >= N) return;
  const int t = (int)(i >> 7);
  const int d = (int)(i & (DIM - 1));
  const float lrs = lr[0] * LR_C;
  const long row = (long)flat[t] * DIM + d;
  atomicAdd(&out_u[row], lrs * gu[i]);
  atomicAdd(&out_v[row], lrs * gv[i]);
}

extern "C" void kernel_launch(void* const* d_in, const int* in_sizes, int n_in,
                              void* d_out, int out_size, void* d_ws, size_t ws_size,
                              hipStream_t stream) {
  const float* U    = (const float*)d_in[0];   // u_weights  [EMB, DIM]
  const float* V    = (const float*)d_in[1];   // v_weights  [EMB, DIM]
  const float* lut  = (const float*)d_in[2];   // lookup_table [1202]
  const int*   flat = (const int*)  d_in[3];   // nodes flattened [TOK]
  const int*   ipu  = (const int*)  d_in[4];   // idx_pos_u [NPOS]
  const int*   ipv  = (const int*)  d_in[5];   // idx_pos_v [NPOS]
  const int*   inu  = (const int*)  d_in[6];   // idx_neg_u [NNEG]
  const int*   inv  = (const int*)  d_in[7];   // idx_neg_v [NNEG]
  const float* lr   = (const float*)d_in[8];   // lr [1]

  float* out_u = (float*)d_out;                    // new_u
  float* out_v = out_u + (long)EMB * DIM;          // new_v
  float* gu = (float*)d_ws;                        // grad_u [TOK, DIM]
  float* gv = gu + (long)TOK * DIM;                // grad_v [TOK, DIM]

  copy_tables<<<2048, 256, 0, stream>>>((const float4*)U, (const float4*)V,
                                        (float4*)d_out);
  zero_grads<<<(TOK * DIM * 2 / 4 + 255) / 256, 256, 0, stream>>>((float4*)d_ws);
  pair_grads<<<(NPOS / 16 + NNEG / 16) / 8, 256, 0, stream>>>(
      U, V, lut, flat, ipu, ipv, inu, inv, gu, gv);
  apply_sgd<<<((long)TOK * DIM + 255) / 256, 256, 0, stream>>>(
      flat, gu, gv, lr, out_u, out_v);
}